// CRFLayer_79010218377499
// MI455X (gfx1250) — compile-verified
//
#include <hip/hip_runtime.h>
#include <cstdint>

// Problem constants from the reference: B=64, C=2, H=W=256 -> T=65536.
#define B_      64
#define T_      65536
#define CHUNK   4096
#define NCHUNK  (T_ / CHUNK)        // 16
#define THREADS 256
#define PER_THR (CHUNK / THREADS)   // 16 consecutive timesteps per thread

// CDNA5 async global->LDS path (ASYNCcnt-tracked), guarded so we always compile.
#if defined(__HIP_DEVICE_COMPILE__) && defined(__gfx1250__) && \
    __has_builtin(__builtin_amdgcn_global_load_async_to_lds_b128)
#define USE_ASYNC_LDS 1
#endif

// Pointee type the builtin expects for b128: int4 (per hipcc diagnostic:
// "__attribute__((__vector_size__(4 * sizeof(int)))) int *").
typedef int v4i_ __attribute__((ext_vector_type(4)));

// logsumexp of two values; -inf-safe (matches jax.nn.logsumexp semantics).
__device__ __forceinline__ float lse2(float a, float b) {
  float mx = fmaxf(a, b);
  float mn = fminf(a, b);
  float r  = mx + log1pf(expf(mn - mx));
  return (mn == -__builtin_inff()) ? mx : r;
}

// Kernel 1: per (batch, chunk) — stage chunk into LDS, write decoded,
// build the ordered log-semiring product of the chunk's 2x2 matrices.
// P layout: [p00, p01, p10, p11]; chain applies newer M on the LEFT.
__global__ __launch_bounds__(THREADS) void crf_chunk_kernel(
    const float* __restrict__ logits,
    const float* __restrict__ trans,
    float* __restrict__ out,
    float* __restrict__ ws) {
  __shared__ __align__(16) float se0[CHUNK];   // emit channel 0
  __shared__ __align__(16) float se1[CHUNK];   // emit channel 1
  __shared__ float pm[THREADS][4];             // tree-reduction matrices

  const int tid        = threadIdx.x;
  const int chunk      = blockIdx.x;
  const int b          = blockIdx.y;
  const int chunkStart = chunk * CHUNK;

  // emits[b,t,c] = logits[(b*2 + c)*T + t]
  const float* g0 = logits + (size_t)b * (2u * T_) + chunkStart;
  const float* g1 = g0 + T_;

#if defined(USE_ASYNC_LDS)
  #pragma unroll
  for (int k = 0; k < PER_THR / 4; ++k) {
    const int vi = (tid + THREADS * k) * 4;   // 16B-aligned, lane-consecutive
    __builtin_amdgcn_global_load_async_to_lds_b128((v4i_*)(g0 + vi), (v4i_*)(&se0[vi]), 0, 0);
    __builtin_amdgcn_global_load_async_to_lds_b128((v4i_*)(g1 + vi), (v4i_*)(&se1[vi]), 0, 0);
  }
  #if __has_builtin(__builtin_amdgcn_s_wait_asynccnt)
  __builtin_amdgcn_s_wait_asynccnt(0);
  #else
  asm volatile("s_wait_asynccnt 0x0" ::: "memory");
  #endif
#else
  #pragma unroll
  for (int k = 0; k < PER_THR / 4; ++k) {
    const int vi = (tid + THREADS * k) * 4;
    const float4 v0 = *reinterpret_cast<const float4*>(g0 + vi);
    const float4 v1 = *reinterpret_cast<const float4*>(g1 + vi);
    *reinterpret_cast<float4*>(&se0[vi]) = v0;
    *reinterpret_cast<float4*>(&se1[vi]) = v1;
  }
#endif
  __syncthreads();

  // decoded = argmax over C=2 (ties -> 0), written as float, coalesced.
  float* dec = out + 1 + (size_t)b * T_ + chunkStart;
  #pragma unroll
  for (int k = 0; k < PER_THR; ++k) {
    const int lt = tid + THREADS * k;
    dec[lt] = (se1[lt] > se0[lt]) ? 1.0f : 0.0f;
  }

  const float tr00 = trans[0], tr01 = trans[1];
  const float tr10 = trans[2], tr11 = trans[3];

  // Ordered product over this thread's 16 consecutive timesteps.
  // Matrices exist only for global t >= 1 (alpha_0 applied in kernel 2).
  const int lt0  = tid * PER_THR;
  const int skip = (chunkStart + lt0 == 0) ? 1 : 0;
  float e0 = se0[lt0 + skip], e1 = se1[lt0 + skip];
  float p00 = tr00 + e0, p01 = tr01 + e0;   // P = M_{t_first}
  float p10 = tr10 + e1, p11 = tr11 + e1;
  for (int k = skip + 1; k < PER_THR; ++k) {
    e0 = se0[lt0 + k]; e1 = se1[lt0 + k];
    const float n00 = lse2(tr00 + p00, tr01 + p10) + e0;   // P <- M_t (x) P
    const float n01 = lse2(tr00 + p01, tr01 + p11) + e0;
    const float n10 = lse2(tr10 + p00, tr11 + p10) + e1;
    const float n11 = lse2(tr10 + p01, tr11 + p11) + e1;
    p00 = n00; p01 = n01; p10 = n10; p11 = n11;
  }

  pm[tid][0] = p00; pm[tid][1] = p01; pm[tid][2] = p10; pm[tid][3] = p11;

  // Ordered tree combine: thread i+s covers LATER t -> goes on the left.
  for (int s = 1; s < THREADS; s <<= 1) {
    __syncthreads();
    if ((tid & (2 * s - 1)) == 0) {
      const float a00 = pm[tid + s][0], a01 = pm[tid + s][1];
      const float a10 = pm[tid + s][2], a11 = pm[tid + s][3];
      const float n00 = lse2(a00 + p00, a01 + p10);
      const float n01 = lse2(a00 + p01, a01 + p11);
      const float n10 = lse2(a10 + p00, a11 + p10);
      const float n11 = lse2(a10 + p01, a11 + p11);
      p00 = n00; p01 = n01; p10 = n10; p11 = n11;
      pm[tid][0] = p00; pm[tid][1] = p01; pm[tid][2] = p10; pm[tid][3] = p11;
    }
  }

  if (tid == 0) {
    float* w = ws + ((size_t)b * NCHUNK + chunk) * 4;
    w[0] = p00; w[1] = p01; w[2] = p10; w[3] = p11;
  }
}

// Kernel 2: one thread per batch folds chunk matrices onto alpha_0 in order;
// fixed-order sum over batches -> deterministic loss at out[0].
__global__ __launch_bounds__(B_) void crf_finalize_kernel(
    const float* __restrict__ logits,
    const float* __restrict__ ws,
    float* __restrict__ out) {
  __shared__ float ll[B_];
  const int b = threadIdx.x;
  float a0 = logits[(size_t)b * (2u * T_)];        // emits[b,0,0]
  float a1 = logits[(size_t)b * (2u * T_) + T_];   // emits[b,0,1]
  const float* w = ws + (size_t)b * NCHUNK * 4;
  for (int c = 0; c < NCHUNK; ++c) {
    const float p00 = w[4 * c + 0], p01 = w[4 * c + 1];
    const float p10 = w[4 * c + 2], p11 = w[4 * c + 3];
    const float n0 = lse2(p00 + a0, p01 + a1);
    const float n1 = lse2(p10 + a0, p11 + a1);
    a0 = n0; a1 = n1;
  }
  ll[b] = lse2(a0, a1);
  __syncthreads();
  if (b == 0) {
    float s = 0.0f;
    for (int i = 0; i < B_; ++i) s += ll[i];   // fixed order: deterministic
    out[0] = -s / (float)B_;
  }
}

extern "C" void kernel_launch(void* const* d_in, const int* in_sizes, int n_in,
                              void* d_out, int out_size, void* d_ws, size_t ws_size,
                              hipStream_t stream) {
  const float* logits = (const float*)d_in[0];
  // d_in[1] (mask) is unused by the reference computation.
  const float* trans  = (const float*)d_in[2];
  float* out = (float*)d_out;   // out[0] = loss, out[1..] = decoded (B*T floats)
  float* ws  = (float*)d_ws;    // B*NCHUNK*4 floats = 16 KB of chunk matrices

  dim3 grid(NCHUNK, B_);
  crf_chunk_kernel<<<grid, THREADS, 0, stream>>>(logits, trans, out, ws);
  crf_finalize_kernel<<<1, B_, 0, stream>>>(logits, ws, out);
}